// SpeechEncoderCTC_2800318677389
// MI455X (gfx1250) — compile-verified
//
#include <hip/hip_runtime.h>
#include <hip/hip_bf16.h>
#include <stdint.h>

// ---------------- problem constants (match reference setup_inputs) ----------
#define T_    1000
#define B_    32
#define C_    1024
#define L_    200
#define S_    401          // 2*L+1
#define NT    416          // 13 wave32's, one thread per s position
#define PF    8            // async prefetch ring depth (power of two)
#define INF_  (-100000.0f)

typedef __attribute__((ext_vector_type(2))) float v2f;
typedef __attribute__((ext_vector_type(8))) float v8f;

// ---------------- CDNA5 helpers ---------------------------------------------

// Per-lane async gather: global -> LDS, tracked by ASYNCcnt.
__device__ __forceinline__ void async_gather_f32(const float* g, float* l) {
#if __has_builtin(__builtin_amdgcn_global_load_async_to_lds_b32)
  __builtin_amdgcn_global_load_async_to_lds_b32(
      (__attribute__((address_space(1))) int*)g,
      (__attribute__((address_space(3))) int*)l,
      /*offset=*/0, /*cpol=*/0);
#else
  unsigned lds_off =
      (unsigned)(unsigned long long)(__attribute__((address_space(3))) float*)l;
  asm volatile("global_load_async_to_lds_b32 %0, %1, off"
               :: "v"(lds_off), "v"(g) : "memory");
#endif
}

template <int N>
__device__ __forceinline__ void wait_asynccnt_le() {
#if __has_builtin(__builtin_amdgcn_s_wait_asynccnt)
  __builtin_amdgcn_s_wait_asynccnt((unsigned short)N);
#else
  asm volatile("s_wait_asynccnt %0" :: "n"(N) : "memory");
#endif
}

// log(exp(a)+exp(b)); exact for equal args, saturates cleanly for far args.
__device__ __forceinline__ float lae(float a, float b) {
  float m = fmaxf(a, b);
  float d = fabsf(a - b);
  return m + log1pf(__expf(-d));
}

// ---------------- kernel 1: per-batch CTC alpha recurrence ------------------
// grid = B_ blocks (one batch per WGP), block = NT threads (thread == s pos).
__global__ __launch_bounds__(NT) void ctc_alpha_kernel(
    const float* __restrict__ log_probs,   // (T,B,C)
    const int*   __restrict__ targets,     // (B,L)
    const int*   __restrict__ in_len,      // (B,)
    const int*   __restrict__ tgt_len,     // (B,)
    float*       __restrict__ loss_out) {  // (B,) per-batch -logp/target_len
  __shared__ float lp_ring[PF][NT];        // async-gathered labels_prob rows
  __shared__ float alpha[2][NT + 2];       // +2 front pad = virtual s=-2,-1

  const int b = blockIdx.x;
  const int s = threadIdx.x;
  const bool active = (s < S_);

  // interleaved label for position s: even -> blank(0), odd -> targets[(s-1)/2]
  int col = 0;
  bool tm = false;                          // skip-transition allowed at s
  if (active && (s & 1)) {
    col = targets[b * L_ + (s >> 1)];
    if (s >= 3) {
      int prevcol = targets[b * L_ + (s >> 1) - 1];
      tm = (col != prevcol);
    }
  }

  const int inlen = in_len[b];              // uniform within block
  const int tl    = tgt_len[b];
  const int Tlim  = inlen;                  // alphas past inlen-1 are unused

  // lane's gather address at t=0; advances by B_*C_ floats per t
  const float* gbase = log_probs + (size_t)b * C_ + col;

  // prologue: fill the ring PF rows deep
  for (int q = 0; q < PF; ++q)
    if (active) async_gather_f32(gbase + (size_t)q * (B_ * C_), &lp_ring[q][s]);

  // virtual alpha[-2], alpha[-1] = INF in both buffers
  if (s < 2) { alpha[0][s] = INF_; alpha[1][s] = INF_; }
  __syncthreads();

  int cur = 0;
  for (int t = 0; t < Tlim; ++t) {
    const int slot = t & (PF - 1);

    // retire the oldest async gather (row t); in-order per-wave completion.
    if (t <= T_ - 1 - PF) wait_asynccnt_le<PF - 1>();
    else                  wait_asynccnt_le<0>();

    const float lp = active ? lp_ring[slot][s] : INF_;

    float na;
    if (t == 0) {
      na = (s < 2) ? lp : INF_;             // alpha0
    } else {
      float a0 = alpha[cur][s + 2];         // alpha_prev[s]
      float a1 = alpha[cur][s + 1];         // alpha_prev[s-1]
      float r  = lae(a0, a1);
      if (tm) r = lae(r, alpha[cur][s]);    // alpha_prev[s-2]
      na = r + lp;
    }
    const int nxt = cur ^ 1;
    if (active) alpha[nxt][s + 2] = na;
    __syncthreads();                        // also drains DScnt before reuse

    // refill the slot we just consumed with row t+PF
    if (active && (t + PF < T_))
      async_gather_f32(gbase + (size_t)(t + PF) * (B_ * C_), &lp_ring[slot][s]);

    cur = nxt;
  }

  // alpha[cur] now holds row t = inlen-1
  if (threadIdx.x == 0) {
    const int s1 = 2 * tl;                  // final blank position (<= 400)
    float v1 = alpha[cur][s1 + 2];
    float v2 = alpha[cur][s1 + 1];          // s2 = s1-1 (tl >= 1 always)
    float sp = lae(v1, v2);
    if (isinf(sp)) sp = 0.0f;               // zero_infinity
    loss_out[b] = -sp / (float)tl;
  }
}

// ---------------- kernel 2: mean over B via V_WMMA_F32_16X16X4_F32 ----------
// losses packed into B-operand (two K-rows carry loss[2n], loss[2n+1] in
// column n); A = all 1/32  =>  D[m,n] = (loss[2n]+loss[2n+1])/32 for every m.
// Robust to the exact K<->VGPR lane mapping since A is uniform and each loss
// occupies a unique (k,n) slot with n = lane. 32-lane shuffle sum counts each
// column twice (M=0 and M=8 halves) -> multiply by 0.5.
__global__ __launch_bounds__(32) void ctc_reduce_kernel(
    const float* __restrict__ loss_in, float* __restrict__ out) {
  const int lane = threadIdx.x;             // exactly one wave32, EXEC all 1s
  const int n = lane & 15;
  float l0 = loss_in[2 * n];
  float l1 = loss_in[2 * n + 1];
  v2f Bv;
  Bv.x = (lane < 16) ? l0 : 0.0f;
  Bv.y = (lane < 16) ? l1 : 0.0f;
  v2f Av;
  Av.x = 1.0f / 32.0f;
  Av.y = 1.0f / 32.0f;
  v8f Cv = {};
  v8f D = __builtin_amdgcn_wmma_f32_16x16x4_f32(
      /*neg_a=*/false, Av, /*neg_b=*/false, Bv,
      /*c_mod=*/(short)0, Cv, /*reuse_a=*/false, /*reuse_b=*/false);
  float r = D[0];
  #pragma unroll
  for (int off = 16; off > 0; off >>= 1) r += __shfl_xor(r, off, 32);
  r *= 0.5f;
  if (lane == 0) *out = r;
}

// ---------------- launch ----------------------------------------------------
extern "C" void kernel_launch(void* const* d_in, const int* in_sizes, int n_in,
                              void* d_out, int out_size, void* d_ws, size_t ws_size,
                              hipStream_t stream) {
  const float* log_probs = (const float*)d_in[0];   // (T,B,C) f32
  const int*   targets   = (const int*)d_in[1];     // (B,L)  i32
  const int*   in_len    = (const int*)d_in[2];     // (B,)   i32
  const int*   tgt_len   = (const int*)d_in[3];     // (B,)   i32
  float* out   = (float*)d_out;                     // scalar
  float* per_b = (float*)d_ws;                      // B_ floats scratch

  ctc_alpha_kernel<<<dim3(B_), dim3(NT), 0, stream>>>(
      log_probs, targets, in_len, tgt_len, per_b);
  ctc_reduce_kernel<<<dim3(1), dim3(32), 0, stream>>>(per_b, out);
}